// StructuredLinear_30846455120727
// MI455X (gfx1250) — compile-verified
//
#include <hip/hip_runtime.h>
#include <stdint.h>

// Problem sizes (fixed by the reference)
#define NDIM  4096   // layer size N
#define RRANK 4      // displacement rank R
#define BATCH 8192   // batch B

typedef __attribute__((ext_vector_type(2))) float v2f;
typedef __attribute__((ext_vector_type(8))) float v8f;

// ---------------------------------------------------------------------------
// Phase 1: build W[n,m] = 0.5*C[d] - S[d,m],  d = (n-m) mod N
//   p_d[j] = sum_r G[(d+j)%N][r] * H[j][r]
//   C[d]   = sum_j p_d[j]
//   S[d,m] = sum_{j>m} p_d[j]   (suffix sum)
// One block per diagonal d. O(N^2 R) total — negligible vs the GEMM.
// ---------------------------------------------------------------------------
__global__ __launch_bounds__(256)
void build_w_kernel(const float* __restrict__ G,
                    const float* __restrict__ H,
                    float* __restrict__ W) {
  const int d   = blockIdx.x;
  const int tid = threadIdx.x;

  __shared__ float p[NDIM];     // 16 KB
  __shared__ float incl[256];   // inclusive suffix sums of 16-elem chunks

  // 1) p[j] = <G[(d+j)%N], H[j]>  (R=4, one float4 per row), coalesced fill
  for (int i = 0; i < 16; ++i) {
    int j  = i * 256 + tid;
    int gi = (d + j) & (NDIM - 1);
    const float4 g = *reinterpret_cast<const float4*>(G + (size_t)gi * RRANK);
    const float4 h = *reinterpret_cast<const float4*>(H + (size_t)j  * RRANK);
    p[j] = g.x * h.x + g.y * h.y + g.z * h.z + g.w * h.w;
  }
  __syncthreads();

  // 2) per-thread chunk sum (chunk = 16 contiguous j)
  float csum = 0.f;
#pragma unroll
  for (int q = 0; q < 16; ++q) csum += p[tid * 16 + q];
  incl[tid] = csum;
  __syncthreads();

  // inclusive suffix scan over the 256 chunk sums (Hillis-Steele, reversed)
  for (int off = 1; off < 256; off <<= 1) {
    float v = (tid + off < 256) ? incl[tid + off] : 0.f;
    __syncthreads();
    incl[tid] += v;
    __syncthreads();
  }
  const float totalC = incl[0];

  // 3) in-chunk suffix pass (right-to-left, in place; thread-private chunk)
  float run = (tid < 255) ? incl[tid + 1] : 0.f;
#pragma unroll
  for (int q = 15; q >= 0; --q) {
    int j = tid * 16 + q;
    float t = p[j];
    p[j] = run;          // now p[m] holds S[d,m]
    run += t;
  }
  __syncthreads();

  // 4) write diagonal d of W (scattered 4B stores; W stays L2-resident)
  const float halfC = 0.5f * totalC;
  for (int i = 0; i < 16; ++i) {
    int m = i * 256 + tid;
    int n = (m + d) & (NDIM - 1);
    W[(size_t)n * NDIM + m] = halfC - p[m];
  }
}

// ---------------------------------------------------------------------------
// Phase 2: Y = X @ W via V_WMMA_F32_16X16X4_F32 (exact f32 semantics).
// Block: 256 threads (8 wave32), 128x128 output tile, BK=16 K-stages,
// double-buffered LDS fed by GLOBAL_LOAD_ASYNC_TO_LDS_B128 (ASYNCcnt).
// Async loads complete in order -> s_wait_asynccnt 4 retires the current
// stage while next-stage loads remain in flight.
// ---------------------------------------------------------------------------
#define BM 128
#define BN 128
#define BK 16
#define XS_STRIDE 20    // BK + 4 pad: 16 M-rows hit distinct banks, 16B aligned
#define WS_STRIDE 136   // BN + 8 pad: half-wave K-split lands on disjoint banks
#define NSTAGE (NDIM / BK)

__device__ __forceinline__ void async_b128(uint32_t lds_off, const float* gptr) {
  // LDS destination address comes from the lds_off VGPR; flat-pointer low 32 bits == LDS offset
  asm volatile("global_load_async_to_lds_b128 %0, %1, off"
               :: "v"(lds_off), "v"(gptr) : "memory");
}

__global__ __launch_bounds__(256)
void gemm_xw_kernel(const float* __restrict__ X,
                    const float* __restrict__ W,
                    float* __restrict__ Y) {
  __shared__ float xs[2][BM * XS_STRIDE]; // 2 x 10240 B
  __shared__ float ws[2][BK * WS_STRIDE]; // 2 x  8704 B

  const int tid   = threadIdx.x;
  const int lane  = tid & 31;
  const int wave  = tid >> 5;
  const int waveM = wave & 1;   // 0..1 -> 64-row slabs
  const int waveN = wave >> 1;  // 0..3 -> 32-col slabs
  const int lhalf = lane >> 4;  // half-wave select (K pairs / M+8 rows)
  const int l16   = lane & 15;

  const int b0 = blockIdx.y * BM;
  const int m0 = blockIdx.x * BN;

  // issue one stage's tiles (4 async b128 per thread: 2 for X, 2 for W)
  auto issue_stage = [&](int buf, int kt) {
#pragma unroll
    for (int i = 0; i < 2; ++i) {        // X tile: 128 rows x 16 K
      int idx = i * 256 + tid;
      int row = idx >> 2;
      int c4  = (idx & 3) * 4;
      uint32_t lo = (uint32_t)(size_t)&xs[buf][row * XS_STRIDE + c4];
      async_b128(lo, X + (size_t)(b0 + row) * NDIM + kt + c4);
    }
#pragma unroll
    for (int i = 0; i < 2; ++i) {        // W tile: 16 K-rows x 128 cols
      int idx = i * 256 + tid;
      int row = idx >> 5;
      int c4  = (idx & 31) * 4;
      uint32_t lo = (uint32_t)(size_t)&ws[buf][row * WS_STRIDE + c4];
      async_b128(lo, W + (size_t)(kt + row) * NDIM + m0 + c4);
    }
  };

  const v8f zacc = {0.f, 0.f, 0.f, 0.f, 0.f, 0.f, 0.f, 0.f};
  v8f acc[4][2];
#pragma unroll
  for (int tm = 0; tm < 4; ++tm)
#pragma unroll
    for (int tn = 0; tn < 2; ++tn) acc[tm][tn] = zacc;

  issue_stage(0, 0);   // prologue

  for (int s = 0; s < NSTAGE; ++s) {
    const int cur = s & 1;
    if (s + 1 < NSTAGE) {
      issue_stage(1 - cur, (s + 1) * BK);
      // current stage's 4 ops retired (in-order), next stage's 4 still in flight
      asm volatile("s_wait_asynccnt 4" ::: "memory");
    } else {
      asm volatile("s_wait_asynccnt 0" ::: "memory");
    }
    __syncthreads();   // all waves' tiles for stage s are in LDS

#pragma unroll
    for (int kk = 0; kk < BK; kk += 4) {
      // A frags (16x4 f32): lanes 0-15 -> K = kk+{0,1}; lanes 16-31 -> K = kk+{2,3}
      v2f afrag[4];
#pragma unroll
      for (int tm = 0; tm < 4; ++tm) {
        int mrow = waveM * 64 + tm * 16 + l16;
        const float* ap = &xs[cur][mrow * XS_STRIDE + kk + lhalf * 2];
        afrag[tm].x = ap[0];
        afrag[tm].y = ap[1];
      }
      // B frags (4x16 f32): same half-wave K split, N striped across lanes
      v2f bfrag[2];
#pragma unroll
      for (int tn = 0; tn < 2; ++tn) {
        int ncol = waveN * 32 + tn * 16 + l16;
        bfrag[tn].x = ws[cur][(kk + lhalf * 2)     * WS_STRIDE + ncol];
        bfrag[tn].y = ws[cur][(kk + lhalf * 2 + 1) * WS_STRIDE + ncol];
      }
#pragma unroll
      for (int tm = 0; tm < 4; ++tm)
#pragma unroll
        for (int tn = 0; tn < 2; ++tn)
          acc[tm][tn] = __builtin_amdgcn_wmma_f32_16x16x4_f32(
              false, afrag[tm], false, bfrag[tn],
              (short)0, acc[tm][tn], false, false);
    }
    __syncthreads();   // reads of buf `cur` done before iteration s+1 overwrites it
  }

  // write-out per ISA C/D layout: lanes 0-15 -> M=v, lanes 16-31 -> M=8+v
#pragma unroll
  for (int tm = 0; tm < 4; ++tm) {
#pragma unroll
    for (int tn = 0; tn < 2; ++tn) {
      int col   = m0 + waveN * 32 + tn * 16 + l16;
      int rbase = b0 + waveM * 64 + tm * 16 + lhalf * 8;
#pragma unroll
      for (int v = 0; v < 8; ++v)
        Y[(size_t)(rbase + v) * NDIM + col] = acc[tm][tn][v];
    }
  }
}

// ---------------------------------------------------------------------------
extern "C" void kernel_launch(void* const* d_in, const int* in_sizes, int n_in,
                              void* d_out, int out_size, void* d_ws, size_t ws_size,
                              hipStream_t stream) {
  (void)in_sizes; (void)n_in; (void)out_size; (void)ws_size;
  const float* x = (const float*)d_in[0];   // (B, N) f32
  const float* G = (const float*)d_in[1];   // (N, R) f32
  const float* H = (const float*)d_in[2];   // (N, R) f32
  float* Y = (float*)d_out;                 // (B, N) f32
  float* W = (float*)d_ws;                  // N*N f32 = 64 MB scratch

  build_w_kernel<<<dim3(NDIM), dim3(256), 0, stream>>>(G, H, W);
  gemm_xw_kernel<<<dim3(NDIM / BN, BATCH / BM), dim3(256), 0, stream>>>(x, W, Y);
}